// SpatialAttentionModule_27857157881995
// MI455X (gfx1250) — compile-verified
//
#include <hip/hip_runtime.h>
#include <math.h>

typedef __attribute__((ext_vector_type(2))) float v2f;
typedef __attribute__((ext_vector_type(4))) float v4f;
typedef __attribute__((ext_vector_type(8))) float v8f;

#define B_  32
#define C_  256
#define H_  64
#define W_  64
#define HW_ (H_ * W_)
#define BHW_ (B_ * HW_)
#define NCHUNK_ 8
#define CPC_ (C_ / NCHUNK_)   // 32 channels per chunk

// ---------------------------------------------------------------------------
// Kernel 1a: partial channel max/sum over a 32-channel chunk.
// 262144 threads (8192 waves) so ~16MB of loads are in flight — enough to
// cover HBM latency*BW (23.3 TB/s * ~700ns). Coalesced 512B/wave/channel.
// ---------------------------------------------------------------------------
__global__ void sam_pool_partial_kernel(const float* __restrict__ x,
                                        float* __restrict__ part_max,
                                        float* __restrict__ part_sum) {
    int idx   = blockIdx.x * blockDim.x + threadIdx.x;   // over NCHUNK*B*HW/4
    int chunk = idx / (BHW_ / 4);                        // uniform per block
    int pid   = idx - chunk * (BHW_ / 4);
    int b     = pid / (HW_ / 4);
    int q     = pid - b * (HW_ / 4);
    int p0    = q * 4;

    const float* base = x + (size_t)b * C_ * HW_ + (size_t)chunk * CPC_ * HW_ + p0;

    v4f v  = *(const v4f*)base;
    v4f mx = v;
    v4f sm = v;
#pragma unroll 8
    for (int ci = 1; ci < CPC_; ++ci) {
        v4f u = *(const v4f*)(base + (size_t)ci * HW_);
        mx.x = fmaxf(mx.x, u.x); mx.y = fmaxf(mx.y, u.y);
        mx.z = fmaxf(mx.z, u.z); mx.w = fmaxf(mx.w, u.w);
        sm += u;
    }
    size_t o = (size_t)chunk * BHW_ + b * HW_ + p0;
    *(v4f*)(part_max + o) = mx;
    *(v4f*)(part_sum + o) = sm;
}

// ---------------------------------------------------------------------------
// Kernel 1b: combine the 8 partials -> final max / mean maps (L2-hot, 8MB).
// ---------------------------------------------------------------------------
__global__ void sam_pool_combine_kernel(const float* __restrict__ part_max,
                                        const float* __restrict__ part_sum,
                                        float* __restrict__ maxp,
                                        float* __restrict__ avgp) {
    int idx = blockIdx.x * blockDim.x + threadIdx.x;     // over B*HW/4
    int o   = idx * 4;
    v4f mx = *(const v4f*)(part_max + o);
    v4f sm = *(const v4f*)(part_sum + o);
#pragma unroll
    for (int chunk = 1; chunk < NCHUNK_; ++chunk) {
        v4f vm = *(const v4f*)(part_max + (size_t)chunk * BHW_ + o);
        v4f vs = *(const v4f*)(part_sum + (size_t)chunk * BHW_ + o);
        mx.x = fmaxf(mx.x, vm.x); mx.y = fmaxf(mx.y, vm.y);
        mx.z = fmaxf(mx.z, vm.z); mx.w = fmaxf(mx.w, vm.w);
        sm += vs;
    }
    *(v4f*)(maxp + o) = mx;
    *(v4f*)(avgp + o) = sm * (1.0f / (float)C_);
}

// ---------------------------------------------------------------------------
// Kernel 2: 7x7 conv (2-in, 1-out, pad 3) + bias + sigmoid via WMMA f32 16x16x4.
// Per wave: GEMM over K = 2*49 taps (padded to 100 -> 25 wmma ops),
// M = 16 consecutive output pixels (one row segment), B = weights replicated
// across the 16 N-columns.  A layout (ISA 7.12.2, 16x4 f32): lane L (row M=L&15)
// holds K = 4*kk + 2*(L>=16) in VGPR0 and +1 in VGPR1; B mirrors the K striping
// so each lane's B value is W[taps at its own K rows].
// ---------------------------------------------------------------------------
__device__ __forceinline__ float sam_im2col(const float* __restrict__ maxp,
                                            const float* __restrict__ avgp,
                                            int b, int y, int x, int k) {
    if (k >= 98) return 0.0f;
    int i   = (k < 49) ? 0 : 1;
    int rem = k - i * 49;
    int ky  = rem / 7;
    int kx  = rem - ky * 7;
    int yy  = y + ky - 3;
    int xx  = x + kx - 3;
    bool inb = ((unsigned)yy < (unsigned)H_) && ((unsigned)xx < (unsigned)W_);
    int yyc = min(max(yy, 0), H_ - 1);
    int xxc = min(max(xx, 0), W_ - 1);
    const float* __restrict__ p = i ? avgp : maxp;
    float v = p[b * HW_ + yyc * W_ + xxc];
    return inb ? v : 0.0f;
}

__global__ void sam_conv_wmma_kernel(const float* __restrict__ Wt,   // 98 taps
                                     const float* __restrict__ bias, // 1
                                     const float* __restrict__ maxp,
                                     const float* __restrict__ avgp,
                                     float* __restrict__ samap) {
    int b    = blockIdx.y;
    int wave = threadIdx.x >> 5;                 // wave32
    int lane = threadIdx.x & 31;
    int grp  = blockIdx.x * (blockDim.x >> 5) + wave;  // 0..255 per batch
    int m0   = grp * 16;                         // 16 consecutive pixels, same row
    int y    = m0 >> 6;
    int xb   = m0 & 63;
    int r    = lane & 15;
    int hi   = lane >> 4;
    int x    = xb + r;

    v8f acc = {};
#pragma unroll 1
    for (int kk = 0; kk < 25; ++kk) {
        int ka = kk * 4 + 2 * hi;
        v2f a, bw;
        a.x  = sam_im2col(maxp, avgp, b, y, x, ka);
        a.y  = sam_im2col(maxp, avgp, b, y, x, ka + 1);
        bw.x = (ka     < 98) ? Wt[ka]     : 0.0f;
        bw.y = (ka + 1 < 98) ? Wt[ka + 1] : 0.0f;
        acc = __builtin_amdgcn_wmma_f32_16x16x4_f32(
            false, a, false, bw, (short)0, acc, false, false);
    }

    // D layout: lane half hi holds rows M = v + 8*hi in acc[v]; all 16 columns
    // are identical (B columns replicated), so lanes with r==0 write the tile.
    if (r == 0) {
        float bs = bias[0];
        int base = b * HW_ + m0 + hi * 8;
#pragma unroll
        for (int v = 0; v < 8; ++v) {
            float t = acc[v] + bs;
            samap[base + v] = 1.0f / (1.0f + expf(-t));
        }
    }
}

// ---------------------------------------------------------------------------
// Kernel 3: out = x * samap (broadcast over channels).  float4 per thread;
// x re-read (largely L2-resident from kernel 1), output stored non-temporal
// so the 128MiB store stream does not evict x from L2.
// ---------------------------------------------------------------------------
__global__ void sam_apply_kernel(const float* __restrict__ x,
                                 const float* __restrict__ samap,
                                 float* __restrict__ out) {
    size_t i4 = (size_t)blockIdx.x * blockDim.x + threadIdx.x; // over N/4
    size_t e0 = i4 * 4;
    int b = (int)(e0 >> 20);          // / (C_*HW_) = 1048576
    int p = (int)(e0 & (HW_ - 1));    // % HW_
    v4f s  = *(const v4f*)(samap + b * HW_ + p);
    v4f xv = *(const v4f*)(x + e0);
    v4f o  = xv * s;
    __builtin_nontemporal_store(o, (v4f*)(out + e0));
}

extern "C" void kernel_launch(void* const* d_in, const int* in_sizes, int n_in,
                              void* d_out, int out_size, void* d_ws, size_t ws_size,
                              hipStream_t stream) {
    const float* x    = (const float*)d_in[0];   // [32,256,64,64]
    const float* Wt   = (const float*)d_in[1];   // [1,2,7,7] = 98 floats
    const float* bias = (const float*)d_in[2];   // [1]
    float* out = (float*)d_out;

    float* part_max = (float*)d_ws;                         // [8][B*HW]  4MB
    float* part_sum = part_max + (size_t)NCHUNK_ * BHW_;    // [8][B*HW]  4MB
    float* maxp     = part_sum + (size_t)NCHUNK_ * BHW_;    // [B*HW]
    float* avgp     = maxp + (size_t)BHW_;                  // [B*HW]
    float* samap    = avgp + (size_t)BHW_;                  // [B*HW]

    // 1a) partial pooling: 8 chunks x 32768 float4 columns -> 1024 blocks
    {
        int threads = 256;
        int blocks  = (NCHUNK_ * BHW_ / 4) / threads;       // 1024
        sam_pool_partial_kernel<<<blocks, threads, 0, stream>>>(x, part_max, part_sum);
    }
    // 1b) combine partials -> 128 blocks
    {
        int threads = 256;
        int blocks  = (BHW_ / 4) / threads;                 // 128
        sam_pool_combine_kernel<<<blocks, threads, 0, stream>>>(part_max, part_sum,
                                                                maxp, avgp);
    }
    // 2) conv + sigmoid via WMMA: 256 tiles/batch, 8 waves/block -> grid (32, 32)
    {
        dim3 grid(32, B_);
        sam_conv_wmma_kernel<<<grid, 256, 0, stream>>>(Wt, bias, maxp, avgp, samap);
    }
    // 3) apply: 8,388,608 float4s -> 32768 blocks
    {
        int threads = 256;
        int blocks  = (int)(((size_t)B_ * C_ * HW_ / 4) / threads); // 32768
        sam_apply_kernel<<<blocks, threads, 0, stream>>>(x, samap, out);
    }
}